// NeuralTS_58514634440980
// MI455X (gfx1250) — compile-verified
//
#include <hip/hip_runtime.h>

// ---------------------------------------------------------------------------
// NeuralTS forward on MI455X (gfx1250, wave32, WMMA).
//
//   U_emb = x_user[2048,100000] @ W_w[6,100000]^T        (memory-bound GEMM)
//   V_emb = H_w[:, x_item]^T                              (gather)
//   out   = MLP(concat(U,V))                              (tiny)
//
// HBM floor: 819MB / 23.3TB/s ~ 35us; intensity ~3 FLOP/byte => memory-bound,
// keep fp32 and ride V_WMMA_F32_16X16X4_F32. Contraction order over users is
// commutative, so K-slots are permuted to let every lane consume one b128
// load per two WMMAs (lo half-wave: users n..n+3, hi half-wave: n+4..n+7).
// B columns j>=6 carry garbage (in-bounds row-0 data) but a matmul has no
// cross-column mixing, so it only lands in C columns we never store -> no
// masking needed in the hot loop.
// ---------------------------------------------------------------------------

typedef float v2f __attribute__((ext_vector_type(2)));
typedef float v4f __attribute__((ext_vector_type(4)));
typedef float v8f __attribute__((ext_vector_type(8)));

#define NUSERS  100000
#define NITEMS  100000
#define EMBK    6
#define BATCH   2048
#define TILE_M  16
#define NCHUNK  2000                    // users per wave-chunk (mult of 16)
#define NSPLITS (NUSERS / NCHUNK)       // 50
#define NTILES  (BATCH / TILE_M)        // 128
#define GEMM_WAVES (NTILES * NSPLITS)   // 6400 waves
#define GEMM_BLOCK 256
#define GEMM_GRID  (GEMM_WAVES * 32 / GEMM_BLOCK)  // 800 blocks

__global__ __launch_bounds__(256) void nts_zero_kernel(float* __restrict__ p, int n) {
    int i = blockIdx.x * blockDim.x + threadIdx.x;
    if (i < n) p[i] = 0.0f;
}

// One wave = one (16-row batch tile) x (2000-user chunk). Accumulates a
// 16x16 fp32 C fragment (cols 0..5 live) over 500 WMMAs, then atomically
// folds into the global U accumulator.
__global__ __launch_bounds__(GEMM_BLOCK) void nts_uemb_gemm_kernel(
    const float* __restrict__ x_user,   // [BATCH, NUSERS] row-major
    const float* __restrict__ W_w,      // [EMBK, NUSERS]  row-major
    float* __restrict__ Uacc)           // [BATCH, EMBK]
{
    const int wid   = (blockIdx.x * GEMM_BLOCK + threadIdx.x) >> 5;
    const int lane  = threadIdx.x & 31;
    const int tile  = wid % NTILES;
    const int split = wid / NTILES;

    const int r  = lane & 15;           // row-in-tile (A) / output col j (B/C)
    const int hi = lane >> 4;           // half-wave owns K-slots 2,3

    // 32-bit element offsets from uniform bases (max 2.05e8 elems < 2^31,
    // byte offset 8.2e8 < 2^31) -> saddr + voffset addressing.
    const int abase = (tile * TILE_M + r) * NUSERS;          // A: this lane's row
    const int bbase = ((r < EMBK) ? r : 0) * NUSERS;         // B: W_w row j (clamped,
                                                             //    j>=6 garbage is benign)
    // Each lane's b128 covers 4 users: lo half n..n+3, hi half n+4..n+7.
    const int kskew = hi ? 4 : 0;

    const int n0 = split * NCHUNK;
    v8f c = {0.0f, 0.0f, 0.0f, 0.0f, 0.0f, 0.0f, 0.0f, 0.0f};

    for (int n = n0; n < n0 + NCHUNK; n += 16) {
        // stream-prefetch this row's A data ~16KB ahead (global_prefetch_b8)
        __builtin_prefetch((const void*)(x_user + abase + n + 4096), 0, 1);
        #pragma unroll
        for (int s = 0; s < 2; ++s) {                       // 8 users per step
            const int off = n + 8 * s + kskew;
            v4f a4 = *(const v4f*)(x_user + abase + off);   // global_load_b128
            v4f b4 = *(const v4f*)(W_w    + bbase + off);   // global_load_b128
            v2f a0 = __builtin_shufflevector(a4, a4, 0, 1);
            v2f a1 = __builtin_shufflevector(a4, a4, 2, 3);
            v2f b0 = __builtin_shufflevector(b4, b4, 0, 1);
            v2f b1 = __builtin_shufflevector(b4, b4, 2, 3);
            // D = A(16x4 f32) * B(4x16 f32) + C(16x16 f32)
            c = __builtin_amdgcn_wmma_f32_16x16x4_f32(
                    false, a0, false, b0, (short)0, c, false, false);
            c = __builtin_amdgcn_wmma_f32_16x16x4_f32(
                    false, a1, false, b1, (short)0, c, false, false);
        }
    }

    // C/D layout: VGPR v, lanes 0-15 -> M=v, lanes 16-31 -> M=v+8; N = lane&15.
    if (r < EMBK) {
        const int rowbase = tile * TILE_M + (hi ? 8 : 0);
        #pragma unroll
        for (int v = 0; v < 8; ++v) {
            atomicAdd(&Uacc[(rowbase + v) * EMBK + r], c[v]);
        }
    }
}

// One thread per batch row: item-embedding gather + 12->16->8->(+1)->1 MLP.
__global__ __launch_bounds__(256) void nts_mlp_kernel(
    const float* __restrict__ Uacc,     // [BATCH, EMBK]
    const int*   __restrict__ x_item,   // [BATCH]
    const float* __restrict__ H_w,      // [EMBK, NITEMS]
    const float* __restrict__ W1,       // [16,12]
    const float* __restrict__ b1,       // [16]
    const float* __restrict__ W2,       // [8,16]
    const float* __restrict__ b2,       // [8]
    const float* __restrict__ W3,       // [1,9]
    float* __restrict__ out)            // [BATCH]
{
    const int b = blockIdx.x * blockDim.x + threadIdx.x;
    if (b >= BATCH) return;

    float z[12];
    #pragma unroll
    for (int k = 0; k < EMBK; ++k) z[k] = Uacc[b * EMBK + k];
    const int item = x_item[b];
    #pragma unroll
    for (int k = 0; k < EMBK; ++k) z[EMBK + k] = H_w[k * NITEMS + item];

    float h1[16];
    #pragma unroll
    for (int i = 0; i < 16; ++i) {
        float acc = b1[i];
        #pragma unroll
        for (int j = 0; j < 12; ++j) acc = fmaf(W1[i * 12 + j], z[j], acc);
        h1[i] = fmaxf(acc, 0.0f);
    }
    float h2[8];
    #pragma unroll
    for (int i = 0; i < 8; ++i) {
        float acc = b2[i];
        #pragma unroll
        for (int j = 0; j < 16; ++j) acc = fmaf(W2[i * 16 + j], h1[j], acc);
        h2[i] = fmaxf(acc, 0.0f);
    }
    float o = W3[8];                      // bias slot (ones column)
    #pragma unroll
    for (int j = 0; j < 8; ++j) o = fmaf(W3[j], h2[j], o);
    out[b] = o;
}

extern "C" void kernel_launch(void* const* d_in, const int* in_sizes, int n_in,
                              void* d_out, int out_size, void* d_ws, size_t ws_size,
                              hipStream_t stream) {
    const float* x_user = (const float*)d_in[0];
    const int*   x_item = (const int*)  d_in[1];
    const float* W_w    = (const float*)d_in[2];
    const float* H_w    = (const float*)d_in[3];
    const float* W1     = (const float*)d_in[4];
    const float* b1     = (const float*)d_in[5];
    const float* W2     = (const float*)d_in[6];
    const float* b2     = (const float*)d_in[7];
    const float* W3     = (const float*)d_in[8];
    float* out  = (float*)d_out;
    float* Uacc = (float*)d_ws;         // BATCH*EMBK fp32 accumulator

    (void)in_sizes; (void)n_in; (void)out_size; (void)ws_size;

    nts_zero_kernel<<<(BATCH * EMBK + 255) / 256, 256, 0, stream>>>(Uacc, BATCH * EMBK);
    nts_uemb_gemm_kernel<<<GEMM_GRID, GEMM_BLOCK, 0, stream>>>(x_user, W_w, Uacc);
    nts_mlp_kernel<<<(BATCH + 255) / 256, 256, 0, stream>>>(
        Uacc, x_item, H_w, W1, b1, W2, b2, W3, out);
}